// SSIMLoss_7610682048600
// MI455X (gfx1250) — compile-verified
//
#include <hip/hip_runtime.h>

typedef float v2f __attribute__((ext_vector_type(2)));
typedef float v8f __attribute__((ext_vector_type(8)));

#define DIM       64
#define TILE      16
#define INP       24              // staged extent: 16 + 2*3 halo, padded to 24 (6 K-chunks of 4)
#define VOLS      64              // B*T*C2F = 2*8*4 volumes
#define NBLOCKS   (VOLS * 64)     // 4x4x4 tiles per volume
#define NTHREADS  256

// LDS partition (in floats)
#define OFF_A     0
#define OFF_B     (OFF_A + INP*INP*INP)            // 13824
#define OFF_F0    (OFF_B + INP*INP*INP)            // 27648
#define OFF_F1    (OFF_F0 + INP*INP*TILE)          // 36864
#define OFF_OUT   (OFF_F1 + INP*TILE*TILE)         // 43008
#define SMEM_FLOATS (OFF_OUT + 5*TILE*TILE*TILE)   // 63488
#define SMEM_BYTES  (SMEM_FLOATS * 4)              // 253952 B < 320 KB WGP LDS

// Normalized 7-tap Gaussian, sigma = 1.5 (branchless band lookup; d = k - m)
__device__ __forceinline__ float gauss_w(int d) {
  float r = 0.0f;
  r = (d == 0 || d == 6) ? 0.03663284536f : r;
  r = (d == 1 || d == 5) ? 0.11128075793f : r;
  r = (d == 2 || d == 4) ? 0.21674532589f : r;
  r = (d == 3)           ? 0.27068214159f : r;
  return r;
}

// D(16x16,f32) = A(16x4,f32) * B(4x16,f32) + C
__device__ __forceinline__ v8f wmma4(v2f a, v2f b, v8f c) {
  return __builtin_amdgcn_wmma_f32_16x16x4_f32(false, a, false, b, (short)0, c, false, false);
}

// Compile-time specialized field value: 0:a 1:b 2:a*a 3:b*b 4:a*b
template <int F>
__device__ __forceinline__ v2f qval(const float* sA, const float* sB, int idx) {
  if (F == 0) { return *(const v2f*)(sA + idx); }
  if (F == 1) { return *(const v2f*)(sB + idx); }
  if (F == 2) { v2f a = *(const v2f*)(sA + idx); return a * a; }
  if (F == 3) { v2f b = *(const v2f*)(sB + idx); return b * b; }
  v2f a = *(const v2f*)(sA + idx);
  v2f b = *(const v2f*)(sB + idx);
  return a * b;
}

// One field: separable 7-tap Gaussian via 3 banded-GEMM passes (all WMMA f32 16x16x4)
template <int F>
__device__ __forceinline__ void filter_field(const float* sA, const float* sB,
                                             float* sF0, float* sF1, float* sOut,
                                             const v2f* afrag, int lane, int wave) {
  const int nlo = lane & 15;
  const int kh  = (lane >> 4) << 1;
  const int mb  = (lane >> 4) << 3;

  // ---- Pass Z: q[x:24][y:24][k:24] -> F0[x:24][y:24][m:16], 36 exact N-tiles ----
  for (int tn = wave; tn < (INP * INP) / 16; tn += 8) {
    int n  = tn * 16 + nlo;
    int xx = n / INP, yy = n % INP;
    v8f acc = {0.f, 0.f, 0.f, 0.f, 0.f, 0.f, 0.f, 0.f};
#pragma unroll
    for (int c = 0; c < 6; ++c) {
      int k   = c * 4 + kh;                 // even -> 8B-aligned pair loads
      int idx = (xx * INP + yy) * INP + k;
      acc = wmma4(afrag[c], qval<F>(sA, sB, idx), acc);
    }
#pragma unroll
    for (int v = 0; v < 8; ++v)
      sF0[(xx * INP + yy) * TILE + mb + v] = acc[v];
  }
  __syncthreads();

  // ---- Pass Y: F0[x:24][k:24][z:16] -> F1[x:24][m:16][z:16], 24 exact N-tiles ----
  for (int tn = wave; tn < (INP * TILE) / 16; tn += 8) {
    int n  = tn * 16 + nlo;
    int xx = n >> 4, zz = n & 15;
    v8f acc = {0.f, 0.f, 0.f, 0.f, 0.f, 0.f, 0.f, 0.f};
#pragma unroll
    for (int c = 0; c < 6; ++c) {
      int k = c * 4 + kh;
      v2f q;
      q.x = sF0[(xx * INP + k) * TILE + zz];
      q.y = sF0[(xx * INP + k + 1) * TILE + zz];
      acc = wmma4(afrag[c], q, acc);
    }
#pragma unroll
    for (int v = 0; v < 8; ++v)
      sF1[(xx * TILE + mb + v) * TILE + zz] = acc[v];
  }
  __syncthreads();

  // ---- Pass X: F1[k:24][y:16][z:16] -> OUT[F][m:16][y:16][z:16], 16 exact N-tiles ----
  for (int tn = wave; tn < (TILE * TILE) / 16; tn += 8) {
    int n  = tn * 16 + nlo;
    int yy = n >> 4, zz = n & 15;
    v8f acc = {0.f, 0.f, 0.f, 0.f, 0.f, 0.f, 0.f, 0.f};
#pragma unroll
    for (int c = 0; c < 6; ++c) {
      int k = c * 4 + kh;
      v2f q;
      q.x = sF1[(k * TILE + yy) * TILE + zz];
      q.y = sF1[((k + 1) * TILE + yy) * TILE + zz];
      acc = wmma4(afrag[c], q, acc);
    }
#pragma unroll
    for (int v = 0; v < 8; ++v)
      sOut[F * (TILE * TILE * TILE) + (mb + v) * (TILE * TILE) + yy * TILE + zz] = acc[v];
  }
  __syncthreads();
}

__global__ void __launch_bounds__(NTHREADS)
ssim3d_wmma_kernel(const float* __restrict__ preds,
                   const float* __restrict__ trues,
                   float* __restrict__ partial)
{
  extern __shared__ float smem[];
  float* sA   = smem + OFF_A;
  float* sB   = smem + OFF_B;
  float* sF0  = smem + OFF_F0;   // [x:24][y:24][mz:16]
  float* sF1  = smem + OFF_F1;   // [x:24][my:16][mz:16]
  float* sOut = smem + OFF_OUT;  // [field:5][mx:16][my:16][mz:16]

  const int tid  = threadIdx.x;
  const int lane = tid & 31;
  const int wave = tid >> 5;

  const int vol = blockIdx.x >> 6;
  const int t   = blockIdx.x & 63;
  const int x0  = ((t >> 4) & 3) * TILE - 3;
  const int y0  = ((t >> 2) & 3) * TILE - 3;
  const int z0  = (t & 3) * TILE - 3;

  const float* img1 = preds + (size_t)vol * (DIM * DIM * DIM);
  const float* img2 = trues + (size_t)vol * (DIM * DIM * DIM);

  // ---- Stage 24^3 halo tiles with clamped indices (replicate padding) ----
  for (int i = tid; i < INP * INP * INP; i += NTHREADS) {
    int z = i % INP, y = (i / INP) % INP, x = i / (INP * INP);
    int gx = min(max(x0 + x, 0), DIM - 1);
    int gy = min(max(y0 + y, 0), DIM - 1);
    int gz = min(max(z0 + z, 0), DIM - 1);
    int g  = (gx * DIM + gy) * DIM + gz;
    sA[i] = img1[g];
    sB[i] = img2[g];
  }
  __syncthreads();

  // ---- Per-lane A fragments: banded Gaussian matrix W[16][24], 6 K-chunks ----
  // A layout (16x4 f32): lanes 0-15 -> M=lane; VGPR0: K={0|2}, VGPR1: K={1|3} by lane half
  const int M  = lane & 15;
  const int kh = (lane >> 4) << 1;
  v2f afrag[6];
#pragma unroll
  for (int c = 0; c < 6; ++c) {
    int k0 = c * 4 + kh;
    afrag[c].x = gauss_w(k0 - M);
    afrag[c].y = gauss_w(k0 + 1 - M);
  }

  // ---- Five fields, fully specialized (no runtime field branches) ----
  filter_field<0>(sA, sB, sF0, sF1, sOut, afrag, lane, wave);  // mu1  source a
  filter_field<1>(sA, sB, sF0, sF1, sOut, afrag, lane, wave);  // mu2  source b
  filter_field<2>(sA, sB, sF0, sF1, sOut, afrag, lane, wave);  // E[a^2]
  filter_field<3>(sA, sB, sF0, sF1, sOut, afrag, lane, wave);  // E[b^2]
  filter_field<4>(sA, sB, sF0, sF1, sOut, afrag, lane, wave);  // E[ab]

  // ---- SSIM combine + block reduction (deterministic) ----
  const float C1 = 1.0e-4f, C2 = 9.0e-4f, EPSV = 1.0e-12f;
  float lsum = 0.0f;
  for (int i = tid; i < TILE * TILE * TILE; i += NTHREADS) {
    float mu1 = sOut[0 * 4096 + i];
    float mu2 = sOut[1 * 4096 + i];
    float e11 = sOut[2 * 4096 + i];
    float e22 = sOut[3 * 4096 + i];
    float e12 = sOut[4 * 4096 + i];
    float mu1s = mu1 * mu1, mu2s = mu2 * mu2, mu12 = mu1 * mu2;
    float num = (2.0f * mu12 + C1) * (2.0f * (e12 - mu12) + C2);
    float den = (mu1s + mu2s + C1) * ((e11 - mu1s) + (e22 - mu2s) + C2);
    lsum += 1.0f - num / (den + EPSV);
  }
  sF0[tid] = lsum;                 // F0 region is free; reuse for reduction
  __syncthreads();
  for (int s = NTHREADS / 2; s > 0; s >>= 1) {
    if (tid < s) sF0[tid] += sF0[tid + s];
    __syncthreads();
  }
  if (tid == 0) partial[blockIdx.x] = sF0[0];
}

__global__ void __launch_bounds__(256)
reduce_partials(const float* __restrict__ partial, float* __restrict__ out)
{
  __shared__ float red[256];
  float s = 0.0f;
  for (int i = threadIdx.x; i < NBLOCKS; i += 256) s += partial[i];  // fixed order
  red[threadIdx.x] = s;
  __syncthreads();
  for (int k = 128; k > 0; k >>= 1) {
    if (threadIdx.x < k) red[threadIdx.x] += red[threadIdx.x + k];
    __syncthreads();
  }
  // final = sum of (1-ssim) over all 64 volumes / 16,777,216  ( == (p_loss+s_loss)/2 )
  if (threadIdx.x == 0) out[0] = red[0] * (1.0f / 16777216.0f);
}

extern "C" void kernel_launch(void* const* d_in, const int* in_sizes, int n_in,
                              void* d_out, int out_size, void* d_ws, size_t ws_size,
                              hipStream_t stream) {
  (void)in_sizes; (void)n_in; (void)out_size; (void)ws_size;
  const float* preds = (const float*)d_in[0];
  const float* trues = (const float*)d_in[1];
  float* partial = (float*)d_ws;  // 4096 floats of scratch

  ssim3d_wmma_kernel<<<NBLOCKS, NTHREADS, SMEM_BYTES, stream>>>(preds, trues, partial);
  reduce_partials<<<1, 256, 0, stream>>>(partial, (float*)d_out);
}